// unext_shift_mlp_51187420234155
// MI455X (gfx1250) — compile-verified
//
#include <hip/hip_runtime.h>
#include <hip/hip_bf16.h>

// ---------------------------------------------------------------------------
// UNeXt shifted-MLP block, fused into 5 passes for MI455X (gfx1250, wave32):
//   1. shift(H) + GEMM(w1,b1)            x   -> out
//   2. dwconv3x3x3(dw1)+bdw1             out -> ws
//   3. shift(W) + GEMM(w2,b2)            ws  -> out
//   4. dwconv3x3x3(dw2)+bdw2             out -> ws
//   5. shift(D) + GEMM(w3,b3) + GELU     ws  -> out
// GEMMs use V_WMMA_F32_16X16X4_F32 (full fp32 matrix path) so numerics match
// the fp32 reference up to reduction order. ~840 MB HBM traffic total.
// ---------------------------------------------------------------------------

typedef __attribute__((ext_vector_type(2))) float v2f;
typedef __attribute__((ext_vector_type(8))) float v8f;

#define C_CH   160
#define NSP    32768      // 32*32*32 spatial positions per batch
#define MTOT   131072     // 4 * NSP total rows
#define MBLK   64         // rows per block (4 waves x 16)
#define LDS_P  164        // LDS row stride (floats): bank-conflict-free & 16B aligned

__device__ __forceinline__ float gelu_exact(float x) {
    return 0.5f * x * (1.0f + erff(x * 0.70710678118654752f));
}

// ---------------------------------------------------------------------------
// Fused shift + GEMM (+ optional exact GELU).
// AXIS: 0 = H (row offset 1024), 1 = W (32), 2 = D (1).
// Shift: channel chunk q (32 channels each) reads source coord (axis - (q-2)),
// zero-filled outside [0,32) -- exactly pad/roll/crop of the reference.
// ---------------------------------------------------------------------------
template<int AXIS, int DOGELU>
__global__ __launch_bounds__(128)
void gemm_shift_kernel(const float* __restrict__ X,
                       const float* __restrict__ Wg,
                       const float* __restrict__ bias,
                       float* __restrict__ Y) {
    __shared__ float As[MBLK][LDS_P];

    const int rowBlock = blockIdx.x * MBLK;
    const int stride   = (AXIS == 0) ? 1024 : (AXIS == 1) ? 32 : 1;

    // ---- cooperative shifted A-tile gather: 64 rows x 40 float4 ----
    const float4* __restrict__ X4 = reinterpret_cast<const float4*>(X);
    for (int it = threadIdx.x; it < MBLK * 40; it += 128) {
        const int r   = it / 40;
        const int cg  = it % 40;                 // float4 group of 4 channels
        const int g   = rowBlock + r;            // global row
        const int nsp = g & (NSP - 1);
        const int h = nsp >> 10, w = (nsp >> 5) & 31, d = nsp & 31;
        const int s = (cg >> 3) - 2;             // shift of this 32-ch chunk
        const int coord = ((AXIS == 0) ? h : (AXIS == 1) ? w : d) - s;
        float4 v = make_float4(0.f, 0.f, 0.f, 0.f);
        if (coord >= 0 && coord < 32) {
            v = X4[(g - s * stride) * 40 + cg];
        }
        *reinterpret_cast<float4*>(&As[r][cg << 2]) = v;
    }
    __syncthreads();

    // ---- wave-level WMMA: each wave computes 16 rows x 160 cols ----
    const int lane    = threadIdx.x & 31;
    const int wave    = threadIdx.x >> 5;    // 0..3
    const int m       = lane & 15;
    const int hi      = lane >> 4;           // 0 = lanes 0-15, 1 = lanes 16-31
    const int khalf   = hi << 1;             // K sub-offset 0 or 2 (ISA 7.12.2)
    const int rowbase = wave << 4;

    v8f acc[10] = {};

    for (int k = 0; k < C_CH; k += 4) {
        // A fragment 16x4 f32: lane m holds (rows rowbase+m, K = k+khalf, +1)
        v2f a;
        a.x = As[rowbase + m][k + khalf];
        a.y = As[rowbase + m][k + khalf + 1];

        const float* __restrict__ wrow = Wg + (k + khalf) * C_CH + m;
        #pragma unroll
        for (int nt = 0; nt < 10; ++nt) {
            // B fragment 4x16 f32 (mirrors A): rows K=k+khalf / k+khalf+1
            v2f b;
            b.x = wrow[nt * 16];
            b.y = wrow[nt * 16 + C_CH];
            acc[nt] = __builtin_amdgcn_wmma_f32_16x16x4_f32(
                false, a, false, b, (short)0, acc[nt], false, false);
        }
    }

    // ---- epilogue: bias (+GELU), C/D layout: VGPR v -> row v + 8*hi ----
    #pragma unroll
    for (int nt = 0; nt < 10; ++nt) {
        const int col = nt * 16 + m;
        const float bj = bias[col];
        #pragma unroll
        for (int v = 0; v < 8; ++v) {
            const int mrow = v + (hi << 3);
            float val = acc[nt][v] + bj;
            if (DOGELU) val = gelu_exact(val);
            Y[(rowBlock + rowbase + mrow) * C_CH + col] = val;
        }
    }
}

// ---------------------------------------------------------------------------
// Depthwise 3x3x3 conv, SAME zero padding, channel-last (B,N,C) layout.
// Thread = one spatial point x 4 channels (float4); channel index fastest
// across threads -> fully coalesced 640B rows. Neighbor reuse via WGP$/L2.
// ---------------------------------------------------------------------------
__global__ __launch_bounds__(256)
void dwconv_kernel(const float* __restrict__ X,
                   const float* __restrict__ Kw,
                   const float* __restrict__ Bi,
                   float* __restrict__ Y) {
    const int gid = blockIdx.x * 256 + threadIdx.x;  // MTOT*40 total
    const int cg  = gid % 40;
    const int sp  = gid / 40;
    const int nsp = sp & (NSP - 1);
    const int h = nsp >> 10, w = (nsp >> 5) & 31, d = nsp & 31;
    const int c0   = cg << 2;
    const int base = sp - nsp;                       // b * NSP

    const float4* __restrict__ X4 = reinterpret_cast<const float4*>(X);
    float ax = 0.f, ay = 0.f, az = 0.f, aw = 0.f;

    #pragma unroll
    for (int i = 0; i < 3; ++i) {
        const int hh = h + i - 1;
        if (hh < 0 || hh >= 32) continue;
        #pragma unroll
        for (int j = 0; j < 3; ++j) {
            const int ww = w + j - 1;
            if (ww < 0 || ww >= 32) continue;
            #pragma unroll
            for (int l = 0; l < 3; ++l) {
                const int dd = d + l - 1;
                if (dd < 0 || dd >= 32) continue;
                const int nn  = base + (hh << 10) + (ww << 5) + dd;
                const float4 v = X4[nn * 40 + cg];
                const int tap = i * 9 + j * 3 + l;
                ax = fmaf(v.x, Kw[(c0 + 0) * 27 + tap], ax);
                ay = fmaf(v.y, Kw[(c0 + 1) * 27 + tap], ay);
                az = fmaf(v.z, Kw[(c0 + 2) * 27 + tap], az);
                aw = fmaf(v.w, Kw[(c0 + 3) * 27 + tap], aw);
            }
        }
    }
    float4 o;
    o.x = ax + Bi[c0 + 0];
    o.y = ay + Bi[c0 + 1];
    o.z = az + Bi[c0 + 2];
    o.w = aw + Bi[c0 + 3];
    reinterpret_cast<float4*>(Y)[sp * 40 + cg] = o;
}

// ---------------------------------------------------------------------------
// Inputs (setup_inputs order): 0:x 1:H 2:W 3:D 4:w1 5:b1 6:dw1 7:bdw1
//                              8:w2 9:b2 10:dw2 11:bdw2 12:w3 13:b3
// ws needs one (4,32768,160) fp32 tensor = 80 MiB (ping-pong with d_out).
// ---------------------------------------------------------------------------
extern "C" void kernel_launch(void* const* d_in, const int* in_sizes, int n_in,
                              void* d_out, int out_size, void* d_ws, size_t ws_size,
                              hipStream_t stream) {
    const float* x    = (const float*)d_in[0];
    const float* w1   = (const float*)d_in[4];
    const float* b1   = (const float*)d_in[5];
    const float* dw1  = (const float*)d_in[6];
    const float* bdw1 = (const float*)d_in[7];
    const float* w2   = (const float*)d_in[8];
    const float* b2   = (const float*)d_in[9];
    const float* dw2  = (const float*)d_in[10];
    const float* bdw2 = (const float*)d_in[11];
    const float* w3   = (const float*)d_in[12];
    const float* b3   = (const float*)d_in[13];

    float* out = (float*)d_out;
    float* tmp = (float*)d_ws;   // 4*32768*160 floats

    const dim3 gGemm(MTOT / MBLK);        // 2048 blocks
    const dim3 bGemm(128);                // 4 waves
    const dim3 gConv(MTOT * 40 / 256);    // 20480 blocks
    const dim3 bConv(256);

    gemm_shift_kernel<0, 0><<<gGemm, bGemm, 0, stream>>>(x,   w1, b1, out);
    dwconv_kernel<<<gConv, bConv, 0, stream>>>(out, dw1, bdw1, tmp);
    gemm_shift_kernel<1, 0><<<gGemm, bGemm, 0, stream>>>(tmp, w2, b2, out);
    dwconv_kernel<<<gConv, bConv, 0, stream>>>(out, dw2, bdw2, tmp);
    gemm_shift_kernel<2, 1><<<gGemm, bGemm, 0, stream>>>(tmp, w3, b3, out);
}